// TrackerFuturePredictor_62130996904201
// MI455X (gfx1250) — compile-verified
//
#include <hip/hip_runtime.h>
#include <math.h>

// Problem constants from the reference
#define BB 4
#define MM 500
#define NN 500
#define CC 91

typedef __attribute__((ext_vector_type(2))) float v2f;
typedef __attribute__((ext_vector_type(8))) float v8f;

__device__ __forceinline__ int imin_i(int a, int b) { return a < b ? a : b; }

// ---------------------------------------------------------------------------
// Kernel 1: sigmoid precompute for both logit tensors (done once; each row is
// re-read by ~32 cost tiles, so we don't want 32x expf recompute).
// ---------------------------------------------------------------------------
__global__ void sigmoid_kernel(const float* __restrict__ p1l,
                               const float* __restrict__ p2l,
                               float* __restrict__ s1,
                               float* __restrict__ s2) {
    int idx = blockIdx.x * blockDim.x + threadIdx.x;
    const int n1 = BB * NN * CC;
    const int n2 = BB * MM * CC;
    if (idx < n1) {
        float x = p1l[idx];
        s1[idx] = 1.0f / (1.0f + expf(-x));
    } else if (idx < n1 + n2) {
        float x = p2l[idx - n1];
        s2[idx - n1] = 1.0f / (1.0f + expf(-x));
    }
}

// ---------------------------------------------------------------------------
// Kernel 2: cost matrix. One wave (32 lanes) per 16x16 output tile.
// Center-distance cross term a.b via V_WMMA_F32_16X16X4_F32 (K=2 padded to 4),
// then cost = 0.5*sqrt(|a|^2+|b|^2-2ab) + 0.5*Linf(sigmoid diffs) with the
// logit tiles staged in LDS.
//
// A 16x4 f32 layout (ISA 7.12.2): lanes 0-15 hold (K=0,K=1) of row M=lane in
// 2 VGPRs; lanes 16-31 hold (K=2,K=3) == our zero pad. B 4x16 mirrors it.
// C/D 16x16 f32: VGPR i, lanes 0-15 -> (M=i, N=lane); lanes 16-31 ->
// (M=i+8, N=lane-16).
// ---------------------------------------------------------------------------
__global__ void cost_kernel(const float* __restrict__ p1b,
                            const float* __restrict__ p2b,
                            const float* __restrict__ s1,
                            const float* __restrict__ s2,
                            float* __restrict__ cost) {
    const int TN = (NN + 15) / 16;            // 32 tiles along N
    const int TM = (MM + 15) / 16;            // 32 tiles along M
    int bid = blockIdx.x;
    int b  = bid / (TM * TN);
    int t  = bid % (TM * TN);
    int m0 = (t / TN) * 16;
    int n0 = (t % TN) * 16;
    int lane = threadIdx.x;

    __shared__ float sq2[16];                 // |center2|^2 per tile row
    __shared__ float sq1[16];                 // |center1|^2 per tile col
    __shared__ float ls2[16 * CC];            // sigmoid(pred2_logits) tile
    __shared__ float ls1[16 * CC];            // sigmoid(pred1_logits) tile

    // WMMA A/B operands: lanes 0-15 carry real K=0,1 data; lanes 16-31 carry
    // the K=2,3 zero pad.
    v2f a  = {0.0f, 0.0f};
    v2f bv = {0.0f, 0.0f};
    if (lane < 16) {
        int m = imin_i(m0 + lane, MM - 1);
        const float* pb = p2b + ((size_t)b * MM + m) * 4;
        float x = pb[0], y = pb[1];
        a.x = x; a.y = y;
        sq2[lane] = x * x + y * y;
        int n = imin_i(n0 + lane, NN - 1);
        const float* qb = p1b + ((size_t)b * NN + n) * 4;
        float bx = qb[0], by = qb[1];
        bv.x = bx; bv.y = by;
        sq1[lane] = bx * bx + by * by;
    }
    // Stage the 16xC logit slices (clamped rows for edge tiles).
    const float* s2base = s2 + (size_t)b * MM * CC;
    const float* s1base = s1 + (size_t)b * NN * CC;
    for (int idx = lane; idx < 16 * CC; idx += 32) {
        int r = idx / CC;
        int c = idx - r * CC;
        ls2[idx] = s2base[(size_t)imin_i(m0 + r, MM - 1) * CC + c];
        ls1[idx] = s1base[(size_t)imin_i(n0 + r, NN - 1) * CC + c];
    }
    __syncthreads();

    // dot[i] = center2[m0+mrow] . center1[n0+ncol]  (f32 WMMA, K=4, 2 padded)
    v8f cacc = {0.f, 0.f, 0.f, 0.f, 0.f, 0.f, 0.f, 0.f};
    v8f dot = __builtin_amdgcn_wmma_f32_16x16x4_f32(
        /*neg_a=*/false, a, /*neg_b=*/false, bv,
        /*c_mod=*/(short)0, cacc, /*reuse_a=*/false, /*reuse_b=*/false);

    int n_loc  = lane & 15;
    int m_base = (lane >> 4) << 3;            // 0 or 8

    // L-inf class disparity for this lane's 8 (m,n) cells.
    float cls[8];
#pragma unroll
    for (int i = 0; i < 8; ++i) cls[i] = 0.0f;
    for (int c = 0; c < CC; ++c) {
        float s1v = ls1[n_loc * CC + c];
#pragma unroll
        for (int i = 0; i < 8; ++i) {
            float d = fabsf(ls2[(m_base + i) * CC + c] - s1v);
            cls[i] = fmaxf(cls[i], d);
        }
    }

#pragma unroll
    for (int i = 0; i < 8; ++i) {
        int m = m0 + m_base + i;
        int n = n0 + n_loc;
        if (m < MM && n < NN) {
            float d2 = sq2[m_base + i] + sq1[n_loc] - 2.0f * dot[i];
            float cd = sqrtf(fmaxf(d2, 0.0f));
            cost[((size_t)b * MM + m) * NN + n] = 0.5f * cd + 0.5f * cls[i];
        }
    }
}

// ---------------------------------------------------------------------------
// Kernel 3: exact LAPJV (shortest augmenting path), one workgroup per batch.
// Faithful port of the reference's float64 algorithm; all state in LDS.
// The per-step argmin (the sequential critical path) is done as a wave32
// cross-lane __shfl_xor reduction (no barriers) + one barrier + a 16-entry
// serial fold by thread 0, instead of a 9-barrier LDS tree.
// ---------------------------------------------------------------------------
#define LTH 512
#define NWAVE (LTH / 32)
__global__ void __launch_bounds__(LTH)
lapjv_kernel(const float* __restrict__ cost, int* __restrict__ mapping) {
    int b   = blockIdx.x;
    int tid = threadIdx.x;
    const float* costB = cost + (size_t)b * MM * NN;
    const double INF = __builtin_inf();

    __shared__ double sh_u[LTH], sh_v[LTH], sh_short[LTH];
    __shared__ int sh_path[LTH], sh_c4r[LTH], sh_r4c[LTH], sh_SR[LTH],
                   sh_rem[LTH];
    __shared__ double wred_v[NWAVE];
    __shared__ int    wred_i[NWAVE];
    __shared__ int s_i, s_sink, s_nsr;
    __shared__ double s_minval;

    if (tid < NN) {
        sh_u[tid] = 0.0; sh_v[tid] = 0.0;
        sh_c4r[tid] = -1; sh_r4c[tid] = -1;
    }
    __syncthreads();

    for (int cur = 0; cur < MM; ++cur) {
        if (tid < NN) { sh_rem[tid] = 1; sh_short[tid] = INF; sh_path[tid] = -1; }
        if (tid == 0) { s_i = cur; s_minval = 0.0; s_sink = -1; s_nsr = 0; }
        __syncthreads();

        while (true) {
            int    ii   = s_i;
            double minv = s_minval;
            if (tid == 0) { sh_SR[s_nsr] = ii; s_nsr = s_nsr + 1; }

            // Masked relaxation + local candidate.
            double cand;
            int    candi = tid;
            if (tid < NN) {
                double r = minv + (double)costB[(size_t)ii * NN + tid]
                                - sh_u[ii] - sh_v[tid];
                if (sh_rem[tid] && r < sh_short[tid]) {
                    sh_short[tid] = r;
                    sh_path[tid]  = ii;
                }
                cand = sh_rem[tid] ? sh_short[tid] : INF;
            } else {
                cand = INF;
            }

            // Wave32 cross-lane argmin (first-index tie-break), no barriers.
#pragma unroll
            for (int off = 16; off > 0; off >>= 1) {
                double ov = __shfl_xor(cand, off);
                int    oi = __shfl_xor(candi, off);
                if (ov < cand || (ov == cand && oi < candi)) {
                    cand = ov; candi = oi;
                }
            }
            int wave = tid >> 5;
            if ((tid & 31) == 0) { wred_v[wave] = cand; wred_i[wave] = candi; }
            __syncthreads();

            if (tid == 0) {
                double bvv = wred_v[0]; int bii = wred_i[0];
#pragma unroll
                for (int w = 1; w < NWAVE; ++w) {
                    double ov = wred_v[w]; int oi = wred_i[w];
                    if (ov < bvv || (ov == bvv && oi < bii)) {
                        bvv = ov; bii = oi;
                    }
                }
                s_minval = bvv;
                sh_rem[bii] = 0;
                if (sh_r4c[bii] == -1) s_sink = bii;
                else                   s_i = sh_r4c[bii];
            }
            __syncthreads();
            if (s_sink != -1) break;
        }

        // Dual variable updates (use pre-augmentation col4row, as reference).
        double minv = s_minval;
        int    nsr  = s_nsr;
        if (tid == 0) {
            sh_u[cur] += minv;
        } else if (tid < nsr) {
            int sr = sh_SR[tid];                 // SR[1:]
            sh_u[sr] += minv - sh_short[sh_c4r[sr]];
        }
        if (tid < NN && !sh_rem[tid]) sh_v[tid] -= minv - sh_short[tid];
        __syncthreads();

        // Augment alternating path (serial, tiny).
        if (tid == 0) {
            int j = s_sink;
            while (true) {
                int i = sh_path[j];
                sh_r4c[j] = i;
                int tmp = sh_c4r[i];
                sh_c4r[i] = j;
                j = tmp;
                if (i == cur) break;
            }
        }
        __syncthreads();
    }
    if (tid < MM) mapping[b * MM + tid] = sh_c4r[tid];
}

// ---------------------------------------------------------------------------
// Kernel 4: extrapolation. One thread per (b, m); square problem => mapping
// always valid, but keep the has-guard for faithfulness.
// ---------------------------------------------------------------------------
__global__ void extrap_kernel(const float* __restrict__ p1b,
                              const float* __restrict__ p1l,
                              const float* __restrict__ p2b,
                              const float* __restrict__ p2l,
                              const float* __restrict__ toff,
                              const int*   __restrict__ mapping,
                              float* __restrict__ out) {
    int idx = blockIdx.x * blockDim.x + threadIdx.x;
    if (idx >= BB * MM) return;
    int b = idx / MM;

    float first  = toff[b * 3 + 1] - toff[b * 3 + 0];
    float second = toff[b * 3 + 2] - toff[b * 3 + 1];
    float factor = second / first;

    int  j   = mapping[idx];
    bool has = (j >= 0);
    int  jc  = has ? j : 0;

    const float* pb2 = p2b + (size_t)idx * 4;
    const float* pb1 = p1b + ((size_t)b * NN + jc) * 4;
    float* ob = out + (size_t)idx * 4;
#pragma unroll
    for (int k = 0; k < 4; ++k) {
        float c1 = has ? pb1[k] : pb2[k];
        float v  = pb2[k] + (pb2[k] - c1) * factor;
        ob[k] = (k >= 2) ? fmaxf(v, 0.0f) : v;
    }

    const float* pl2 = p2l + (size_t)idx * CC;
    const float* pl1 = p1l + ((size_t)b * NN + jc) * CC;
    float* ol = out + (size_t)BB * MM * 4 + (size_t)idx * CC;
    for (int c = 0; c < CC; ++c) {
        float c1 = has ? pl1[c] : 0.0f;
        ol[c] = 0.5f * (pl2[c] + c1);
    }
}

// ---------------------------------------------------------------------------
extern "C" void kernel_launch(void* const* d_in, const int* in_sizes, int n_in,
                              void* d_out, int out_size, void* d_ws, size_t ws_size,
                              hipStream_t stream) {
    (void)in_sizes; (void)n_in; (void)out_size; (void)ws_size;
    const float* p1b  = (const float*)d_in[0];   // (B,N,4)
    const float* p1l  = (const float*)d_in[1];   // (B,N,C)
    const float* p2b  = (const float*)d_in[2];   // (B,M,4)
    const float* p2l  = (const float*)d_in[3];   // (B,M,C)
    const float* toff = (const float*)d_in[4];   // (B,L)

    // Workspace layout (floats): s1 | s2 | cost | mapping(int)
    float* ws   = (float*)d_ws;
    float* s1   = ws;                                    // B*N*C
    float* s2   = s1 + (size_t)BB * NN * CC;             // B*M*C
    float* cost = s2 + (size_t)BB * MM * CC;             // B*M*N
    int*   map  = (int*)(cost + (size_t)BB * MM * NN);   // B*M
    float* out  = (float*)d_out;                         // boxes3 | logits3

    int tot = BB * NN * CC + BB * MM * CC;
    sigmoid_kernel<<<(tot + 255) / 256, 256, 0, stream>>>(p1l, p2l, s1, s2);

    const int TM = (MM + 15) / 16, TN = (NN + 15) / 16;
    cost_kernel<<<BB * TM * TN, 32, 0, stream>>>(p1b, p2b, s1, s2, cost);

    lapjv_kernel<<<BB, LTH, 0, stream>>>(cost, map);

    int tot2 = BB * MM;
    extrap_kernel<<<(tot2 + 255) / 256, 256, 0, stream>>>(p1b, p1l, p2b, p2l,
                                                          toff, map, out);
}